// DiffusionCoordinateInitializer_45346264711614
// MI455X (gfx1250) — compile-verified
//
#include <hip/hip_runtime.h>

typedef __attribute__((ext_vector_type(2))) float v2f;
typedef __attribute__((ext_vector_type(8))) float v8f;

#define K_DIM 1024
#define N_OUT 3
#define WAVES_PER_BLOCK 8
#define ROWS_PER_WAVE 16
#define ROWS_PER_BLOCK (WAVES_PER_BLOCK * ROWS_PER_WAVE)
#define KPAIRS (K_DIM / 2)

// Fused: target = latent @ W + b  via WMMA f32 16x16x4 with
//   A = zero-padded W (M = output col, LDS, ds_load_2addr_b64, no masking)
//   B = latent rows   (N = row; per-lane global_load_b64 at half*2 lands
//                      each lane exactly on its WMMA K-pair — no shuffles)
// then out = P*noise + (1-P)*target, P = prod_{t=1..steps} t/steps
// (exact closed form of the linear reverse-diffusion scan).
__global__ __launch_bounds__(256) void diffusion_coord_init_kernel(
    const float* __restrict__ latent,   // [rows, 1024]
    const float* __restrict__ W,        // [1024, 3]
    const float* __restrict__ bias,     // [3]
    const float* __restrict__ noise,    // [rows, 3]
    const int*   __restrict__ steps_p,  // [1]
    float*       __restrict__ out,      // [rows, 3]
    int rows)
{
    // Wp[kp*16 + col] = { W[2kp][col], W[2kp+1][col] }, cols 3..15 zero. 64 KB.
    __shared__ v2f Wp[KPAIRS * 16];
    for (int i = threadIdx.x; i < KPAIRS * 16; i += blockDim.x) {
        const int kp = i >> 4, col = i & 15;
        v2f v; v.x = 0.0f; v.y = 0.0f;
        if (col < N_OUT) {
            v.x = W[(2 * kp    ) * N_OUT + col];
            v.y = W[(2 * kp + 1) * N_OUT + col];
        }
        Wp[i] = v;
    }
    __syncthreads();

    const int wave = threadIdx.x >> 5;
    const int lane = threadIdx.x & 31;
    const int half = lane >> 4;     // lane halves carry K pairs (A & B layouts)
    const int l16  = lane & 15;     // A: M (col)   B: N (row)   D: N (row)
    const int row0 = (blockIdx.x * WAVES_PER_BLOCK + wave) * ROWS_PER_WAVE;

    int row = row0 + l16;                 // every lane streams one row
    if (row >= rows) row = rows - 1;      // clamp: WMMA needs EXEC all-ones
    // B layout: lane<16 needs (k, k+1), lane>=16 needs (k+2, k+3) of its row.
    const float* bptr = latent + (size_t)row * K_DIM + half * 2;
    const v2f*   aptr = Wp + half * 16 + l16;

    const float b0 = bias[0], b1 = bias[1], b2 = bias[2];
    v8f c0 = {};
    v8f c1 = {};
    if (half == 0) { c0[0] = b0; c0[1] = b1; c0[2] = b2; }  // seed bias once

    #pragma unroll 4
    for (int k = 0; k < K_DIM; k += 8) {
        const v2f B1 = *(const v2f*)(bptr + k);        // global_load_b64
        const v2f B2 = *(const v2f*)(bptr + k + 4);    // global_load_b64
        const v2f A1 = aptr[((k >> 1)     ) * 16];     // merged ds_load_2addr_b64
        const v2f A2 = aptr[((k >> 1) + 2 ) * 16];

        c0 = __builtin_amdgcn_wmma_f32_16x16x4_f32(
                 false, A1, false, B1, (short)0, c0, false, false);
        c1 = __builtin_amdgcn_wmma_f32_16x16x4_f32(
                 false, A2, false, B2, (short)0, c1, false, false);
    }

    // Closed-form collapse of the 50-step linear scan. One divide, then
    // 2 mults/step (scalarizable) instead of a v_div_scale chain per step.
    const int steps = steps_p[0];
    const float inv = 1.0f / (float)steps;
    float P = 1.0f;
    for (int t = steps; t >= 1; --t) P *= (float)t * inv;
    const float Q = 1.0f - P;

    // D layout: lanes 0-15 hold row = row0+l16, cols 0..7 in c[0..7].
    if (half == 0) {
        const int r = row0 + l16;
        if (r < rows) {
            const int idx = r * N_OUT;
            #pragma unroll
            for (int n = 0; n < N_OUT; ++n)
                out[idx + n] = P * noise[idx + n] + Q * (c0[n] + c1[n]);
        }
    }
}

extern "C" void kernel_launch(void* const* d_in, const int* in_sizes, int n_in,
                              void* d_out, int out_size, void* d_ws, size_t ws_size,
                              hipStream_t stream) {
    const float* latent = (const float*)d_in[0];  // [4*8192, 1024]
    const float* W      = (const float*)d_in[1];  // [1024, 3]
    const float* bias   = (const float*)d_in[2];  // [3]
    const float* noise  = (const float*)d_in[3];  // [4*8192, 3]
    const int*   steps  = (const int*)  d_in[4];  // [1]
    float*       out    = (float*)d_out;          // [4*8192, 3]

    const int rows = in_sizes[0] / K_DIM;         // 32768
    const int grid = (rows + ROWS_PER_BLOCK - 1) / ROWS_PER_BLOCK;  // 256

    diffusion_coord_init_kernel<<<grid, 256, 0, stream>>>(
        latent, W, bias, noise, steps, out, rows);
}